// ChebNet_36498632082159
// MI455X (gfx1250) — compile-verified
//
#include <hip/hip_runtime.h>
#include <hip/hip_bf16.h>

#define NN 100000
#define EE 1600000
#define INF_ 64
#define HID_ 16
#define OUT_ 40

typedef __attribute__((ext_vector_type(2))) float v2f;
typedef __attribute__((ext_vector_type(8))) float v8f;

__device__ __forceinline__ v8f wmma4(v2f a, v2f b, v8f c) {
    // D(16x16,f32) = A(16x4,f32) * B(4x16,f32) + C ; full fp32 precision
    return __builtin_amdgcn_wmma_f32_16x16x4_f32(
        /*neg_a=*/false, a, /*neg_b=*/false, b,
        /*c_mod=*/(short)0, c, /*reuse_a=*/false, /*reuse_b=*/false);
}

// ---------------- graph preprocessing ----------------

__global__ void deg_kernel(const int* __restrict__ row, float* __restrict__ deg) {
    int e = blockIdx.x * blockDim.x + threadIdx.x;
    if (e < EE) atomicAdd(&deg[row[e]], 1.0f);
}

__global__ void dis_kernel(const float* __restrict__ deg, float* __restrict__ dis) {
    int n = blockIdx.x * blockDim.x + threadIdx.x;
    if (n < NN) {
        float d = deg[n];
        dis[n] = (d > 0.0f) ? rsqrtf(fmaxf(d, 1.0f)) : 0.0f;
    }
}

__global__ void norm_kernel(const int* __restrict__ row, const int* __restrict__ col,
                            const float* __restrict__ dis, float* __restrict__ norm) {
    int e = blockIdx.x * blockDim.x + threadIdx.x;
    if (e < EE) norm[e] = -(dis[row[e]] * dis[col[e]]);
}

// ---------------- sparse Lhat: Y[col[e]] += norm[e] * X[row[e]] ----------------
// One thread per (edge, 4-feature chunk): float4 gather + 4 fp32 global atomics.

template <int F>
__global__ void spmm_scatter(const int* __restrict__ row, const int* __restrict__ col,
                             const float* __restrict__ norm,
                             const float* __restrict__ X, float* __restrict__ Y) {
    constexpr int CH = F / 4;                 // chunks per edge (power of two)
    int gid = blockIdx.x * blockDim.x + threadIdx.x;
    int e = gid / CH;
    if (e >= EE) return;
    int c = gid & (CH - 1);
    int r = row[e], d = col[e];
    float w = norm[e];
    const float4 xv = *(const float4*)(X + (size_t)r * F + c * 4);
    float* yp = Y + (size_t)d * F + c * 4;
    atomicAdd(yp + 0, w * xv.x);
    atomicAdd(yp + 1, w * xv.y);
    atomicAdd(yp + 2, w * xv.z);
    atomicAdd(yp + 3, w * xv.w);
}

// ---------------- layer 1 GEMM: h = relu(x@W0 + t1@W1 + (2s - x)@W2 + b) ----------------
// One wave per 16-node tile; 48 v_wmma_f32_16x16x4_f32 per tile; W1 staged in LDS
// and hoisted by the compiler into B-fragment registers across the tile loop.

__global__ void __launch_bounds__(128, 1)
cheb_gemm1(const float* __restrict__ x,   // N x 64 (Tx0)
           const float* __restrict__ t1,  // N x 64 (Tx1)
           const float* __restrict__ s,   // N x 64 (Lhat(Tx1))
           const float* __restrict__ W,   // 3 x 64 x 16
           const float* __restrict__ bias,// 16
           float* __restrict__ h) {       // N x 16
    __shared__ float Wl[3 * 64 * 16];
    for (int i = threadIdx.x; i < 3 * 64 * 16; i += blockDim.x) Wl[i] = W[i];
    __syncthreads();

    const int lane  = threadIdx.x & 31;
    const int wave  = threadIdx.x >> 5;
    const int wpb   = blockDim.x >> 5;
    const int mrow  = lane & 15;            // A: row, B: column (N), within tile
    const int khalf = (lane >> 4) << 1;     // lanes 16-31 take K+2
    const int ntiles = NN / 16;             // 6250 exact

    for (int tile = blockIdx.x * wpb + wave; tile < ntiles; tile += gridDim.x * wpb) {
        const int node0 = tile * 16;
        const float* xr = x  + (size_t)(node0 + mrow) * 64;
        const float* tr = t1 + (size_t)(node0 + mrow) * 64;
        const float* sr = s  + (size_t)(node0 + mrow) * 64;
        v8f acc = {};
#pragma unroll
        for (int i = 0; i < 16; ++i) {
            const int kb = i * 4 + khalf;
            v2f a0 = *(const v2f*)(xr + kb);
            v2f a1 = *(const v2f*)(tr + kb);
            v2f sv = *(const v2f*)(sr + kb);
            v2f a2 = 2.0f * sv - a0;        // Tx2 = 2*Lhat(Tx1) - Tx0, fused
            v2f b0, b1v, b2v;
            b0.x  = Wl[(0 * 64 + kb)     * 16 + mrow];
            b0.y  = Wl[(0 * 64 + kb + 1) * 16 + mrow];
            b1v.x = Wl[(1 * 64 + kb)     * 16 + mrow];
            b1v.y = Wl[(1 * 64 + kb + 1) * 16 + mrow];
            b2v.x = Wl[(2 * 64 + kb)     * 16 + mrow];
            b2v.y = Wl[(2 * 64 + kb + 1) * 16 + mrow];
            acc = wmma4(a0, b0, acc);
            acc = wmma4(a1, b1v, acc);
            acc = wmma4(a2, b2v, acc);
        }
        // C/D layout: VGPR j -> row M = j + 8*(lane>=16); column N = lane&15
        const int ncol  = mrow;
        const int mbase = (lane >> 4) * 8;
        const float bb  = bias[ncol];
#pragma unroll
        for (int j = 0; j < 8; ++j) {
            float v = acc[j] + bb;
            h[(size_t)(node0 + mbase + j) * 16 + ncol] = fmaxf(v, 0.0f);
        }
    }
}

// ---------------- layer 2 GEMM: out = h@W0 + t1@W1 + (2s - h)@W2 + b  (N x 40) ----------------
// OUT=40 padded to 3 column tiles of 16 (zero-padded weights in LDS).

__global__ void __launch_bounds__(128, 1)
cheb_gemm2(const float* __restrict__ h,   // N x 16 (Tx0)
           const float* __restrict__ t1,  // N x 16 (Tx1)
           const float* __restrict__ s,   // N x 16 (Lhat(Tx1))
           const float* __restrict__ W,   // 3 x 16 x 40
           const float* __restrict__ bias,// 40
           float* __restrict__ out) {     // N x 40
    __shared__ float Wl[3 * 16 * 48];
    __shared__ float bl[48];
    for (int i = threadIdx.x; i < 3 * 16 * 48; i += blockDim.x) {
        int n  = i % 48;
        int kr = i / 48;                    // k*16 + kk
        Wl[i] = (n < 40) ? W[kr * 40 + n] : 0.0f;
    }
    for (int i = threadIdx.x; i < 48; i += blockDim.x)
        bl[i] = (i < 40) ? bias[i] : 0.0f;
    __syncthreads();

    const int lane  = threadIdx.x & 31;
    const int wave  = threadIdx.x >> 5;
    const int wpb   = blockDim.x >> 5;
    const int mrow  = lane & 15;
    const int khalf = (lane >> 4) << 1;
    const int ntiles = NN / 16;

    auto ldb = [&](int k, int kb, int c) {
        v2f b;
        b.x = Wl[(k * 16 + kb)     * 48 + c * 16 + mrow];
        b.y = Wl[(k * 16 + kb + 1) * 48 + c * 16 + mrow];
        return b;
    };

    for (int tile = blockIdx.x * wpb + wave; tile < ntiles; tile += gridDim.x * wpb) {
        const int node0 = tile * 16;
        const float* hr = h  + (size_t)(node0 + mrow) * 16;
        const float* tr = t1 + (size_t)(node0 + mrow) * 16;
        const float* sr = s  + (size_t)(node0 + mrow) * 16;
        v8f acc0 = {}, acc1 = {}, acc2 = {};
#pragma unroll
        for (int i = 0; i < 4; ++i) {
            const int kb = i * 4 + khalf;
            v2f a0 = *(const v2f*)(hr + kb);
            v2f a1 = *(const v2f*)(tr + kb);
            v2f sv = *(const v2f*)(sr + kb);
            v2f a2 = 2.0f * sv - a0;
            acc0 = wmma4(a0, ldb(0, kb, 0), acc0);
            acc0 = wmma4(a1, ldb(1, kb, 0), acc0);
            acc0 = wmma4(a2, ldb(2, kb, 0), acc0);
            acc1 = wmma4(a0, ldb(0, kb, 1), acc1);
            acc1 = wmma4(a1, ldb(1, kb, 1), acc1);
            acc1 = wmma4(a2, ldb(2, kb, 1), acc1);
            acc2 = wmma4(a0, ldb(0, kb, 2), acc2);
            acc2 = wmma4(a1, ldb(1, kb, 2), acc2);
            acc2 = wmma4(a2, ldb(2, kb, 2), acc2);
        }
        const int ncol  = mrow;
        const int mbase = (lane >> 4) * 8;
#pragma unroll
        for (int j = 0; j < 8; ++j) {
            const size_t node = (size_t)(node0 + mbase + j);
            out[node * 40 + ncol]      = acc0[j] + bl[ncol];
            out[node * 40 + 16 + ncol] = acc1[j] + bl[16 + ncol];
            if (ncol < 8)
                out[node * 40 + 32 + ncol] = acc2[j] + bl[32 + ncol];
        }
    }
}

// ---------------- row-wise log_softmax over 40 columns, in place ----------------

__global__ void logsoftmax_kernel(float* __restrict__ out) {
    int n = blockIdx.x * blockDim.x + threadIdx.x;
    if (n >= NN) return;
    float* o = out + (size_t)n * 40;
    float m = -INFINITY;
    for (int f = 0; f < 40; ++f) m = fmaxf(m, o[f]);
    float ssum = 0.0f;
    for (int f = 0; f < 40; ++f) ssum += expf(o[f] - m);
    float l = logf(ssum);
    for (int f = 0; f < 40; ++f) o[f] = o[f] - m - l;
}

extern "C" void kernel_launch(void* const* d_in, const int* in_sizes, int n_in,
                              void* d_out, int out_size, void* d_ws, size_t ws_size,
                              hipStream_t stream) {
    const float* x  = (const float*)d_in[0];   // N x 64
    const int*   ei = (const int*)d_in[1];     // 2 x E
    const float* W1 = (const float*)d_in[2];   // 3 x 64 x 16
    const float* b1 = (const float*)d_in[3];   // 16
    const float* W2 = (const float*)d_in[4];   // 3 x 16 x 40
    const float* b2 = (const float*)d_in[5];   // 40
    const int* row = ei;
    const int* col = ei + EE;
    float* out = (float*)d_out;

    // workspace layout (floats): norm[E] | dis[N] | bufA[N*64] | bufB[N*64] | hbuf[N*16]
    float* ws   = (float*)d_ws;
    float* norm = ws;
    float* dis  = norm + EE;
    float* bufA = dis + NN;
    float* bufB = bufA + (size_t)NN * 64;
    float* hbuf = bufB + (size_t)NN * 64;

    const int B = 256;

    // degree (reuse bufA's first N floats), dis, norm
    hipMemsetAsync(bufA, 0, (size_t)NN * sizeof(float), stream);
    deg_kernel<<<(EE + B - 1) / B, B, 0, stream>>>(row, bufA);
    dis_kernel<<<(NN + B - 1) / B, B, 0, stream>>>(bufA, dis);
    norm_kernel<<<(EE + B - 1) / B, B, 0, stream>>>(row, col, dis, norm);

    // layer 1: Tx1 = Lhat(x), S = Lhat(Tx1)
    hipMemsetAsync(bufA, 0, (size_t)NN * 64 * sizeof(float), stream);
    spmm_scatter<64><<<((size_t)EE * 16 + B - 1) / B, B, 0, stream>>>(row, col, norm, x, bufA);
    hipMemsetAsync(bufB, 0, (size_t)NN * 64 * sizeof(float), stream);
    spmm_scatter<64><<<((size_t)EE * 16 + B - 1) / B, B, 0, stream>>>(row, col, norm, bufA, bufB);
    cheb_gemm1<<<200, 128, 0, stream>>>(x, bufA, bufB, W1, b1, hbuf);

    // layer 2: Tx1' = Lhat(h), S' = Lhat(Tx1')  (reuse bufA/bufB at width 16)
    hipMemsetAsync(bufA, 0, (size_t)NN * 16 * sizeof(float), stream);
    spmm_scatter<16><<<((size_t)EE * 4 + B - 1) / B, B, 0, stream>>>(row, col, norm, hbuf, bufA);
    hipMemsetAsync(bufB, 0, (size_t)NN * 16 * sizeof(float), stream);
    spmm_scatter<16><<<((size_t)EE * 4 + B - 1) / B, B, 0, stream>>>(row, col, norm, bufA, bufB);
    cheb_gemm2<<<200, 128, 0, stream>>>(hbuf, bufA, bufB, W2, b2, out);

    logsoftmax_kernel<<<(NN + B - 1) / B, B, 0, stream>>>(out);
}